// AffineTransforms_60455959658580
// MI455X (gfx1250) — compile-verified
//
#include <hip/hip_runtime.h>

typedef __attribute__((ext_vector_type(2))) float v2f;
typedef __attribute__((ext_vector_type(8))) float v8f;

#define HH 384
#define WW 384
#define CC 3

// One wave per 16x16 output tile. Two V_WMMA_F32_16X16X4_F32 ops produce the
// full tile's (ix, iy) sample coordinates directly in the C/D register layout
// (VGPR r: lanes 0-15 -> M=r,N=lane ; lanes 16-31 -> M=r+8,N=lane-16), which
// matches coalesced row-major stores (consecutive lanes = consecutive cols).
__global__ __launch_bounds__(128) void AffineTransforms_wmma_kernel(
    const float* __restrict__ imgs,   // [64,3,384,384]
    const float* __restrict__ theta,  // [64,2,3]
    float* __restrict__ out)          // [64,3,384,384]
{
  const int lane   = threadIdx.x & 31;
  const int wave   = threadIdx.x >> 5;
  const int b      = blockIdx.z;            // batch
  const int tile_y = blockIdx.y;            // 0..23
  const int tile_x = blockIdx.x * 4 + wave; // 0..23

  // Uniform (scalar) theta loads.
  const float t00 = theta[b * 6 + 0], t01 = theta[b * 6 + 1], t02 = theta[b * 6 + 2];
  const float t10 = theta[b * 6 + 3], t11 = theta[b * 6 + 4], t12 = theta[b * 6 + 5];
  // ix = t00*(j-191.5) + t01*(i-191.5) + t02 + 191.5  (the /192,*192 of the
  // reference cancel algebraically); likewise iy.
  const float cx = t02 + 191.5f;
  const float cy = t12 + 191.5f;

  const int  nlo    = lane & 15;
  const bool lohalf = lane < 16;
  const float ylane = (float)(tile_y * 16 + nlo) - 191.5f; // A rows: y_i
  const float xlane = (float)(tile_x * 16 + nlo) - 191.5f; // B cols: x_j

  // A (16x4 f32): lanes 0-15 hold M=lane with VGPR0=K0, VGPR1=K1; K2,K3 (lanes
  // 16-31) are zero. B (4x16): symmetric layout, rows K0,K1 in lanes 0-15.
  v2f a, bx, by;
  a.x  = lohalf ? ylane : 0.0f;                 // K0 row of A = y_i
  a.y  = lohalf ? 1.0f  : 0.0f;                 // K1 row of A = 1
  bx.x = lohalf ? t01   : 0.0f;                 // B row K0 = t01 (broadcast)
  bx.y = lohalf ? (t00 * xlane + cx) : 0.0f;    // B row K1 = t00*x_j + cx
  by.x = lohalf ? t11   : 0.0f;
  by.y = lohalf ? (t10 * xlane + cy) : 0.0f;

  v8f z = {};
  // EXEC is all-1s here (no divergence; full blocks). D = A x B + 0.
  v8f dix = __builtin_amdgcn_wmma_f32_16x16x4_f32(
      false, a, false, bx, (short)0, z, false, false);
  v8f diy = __builtin_amdgcn_wmma_f32_16x16x4_f32(
      false, a, false, by, (short)0, z, false, false);

  const int col      = tile_x * 16 + nlo;
  const int row_base = tile_y * 16 + ((lane >> 4) << 3); // +8 for upper half

  const int    chanStride = HH * WW;
  const float* img0 = imgs + (size_t)b * CC * chanStride;
  float*       outb = out  + (size_t)b * CC * chanStride;

#pragma unroll
  for (int r = 0; r < 8; ++r) {
    const float ix  = dix[r];
    const float iy  = diy[r];
    const float x0f = floorf(ix);
    const float y0f = floorf(iy);
    const float wx1 = ix - x0f, wy1 = iy - y0f;
    const float wx0 = 1.0f - wx1, wy0 = 1.0f - wy1;
    const int xi = (int)x0f, yi = (int)y0f;

    // Fold validity into the weights; clamped indices keep loads in-bounds.
    const bool vx0 = (unsigned)xi       < (unsigned)WW;
    const bool vx1 = (unsigned)(xi + 1) < (unsigned)WW;
    const bool vy0 = (unsigned)yi       < (unsigned)HH;
    const bool vy1 = (unsigned)(yi + 1) < (unsigned)HH;
    const int xc0 = min(max(xi, 0), WW - 1);
    const int xc1 = min(max(xi + 1, 0), WW - 1);
    const int yc0 = min(max(yi, 0), HH - 1);
    const int yc1 = min(max(yi + 1, 0), HH - 1);

    const int o00 = yc0 * WW + xc0;
    const int o10 = yc0 * WW + xc1;
    const int o01 = yc1 * WW + xc0;
    const int o11 = yc1 * WW + xc1;

    const float w00 = (vx0 && vy0) ? wx0 * wy0 : 0.0f;
    const float w10 = (vx1 && vy0) ? wx1 * wy0 : 0.0f;
    const float w01 = (vx0 && vy1) ? wx0 * wy1 : 0.0f;
    const float w11 = (vx1 && vy1) ? wx1 * wy1 : 0.0f;

    const int oout = (row_base + r) * WW + col;

#pragma unroll
    for (int c = 0; c < CC; ++c) {
      const float* img = img0 + c * chanStride;
      const float v00 = img[o00];
      const float v10 = img[o10];
      const float v01 = img[o01];
      const float v11 = img[o11];
      const float res = v00 * w00 + v10 * w10 + v01 * w01 + v11 * w11;
      // Streaming store: output is never re-read; keep L2 for the gathers.
      __builtin_nontemporal_store(res, outb + c * chanStride + oout);
    }
  }
}

extern "C" void kernel_launch(void* const* d_in, const int* in_sizes, int n_in,
                              void* d_out, int out_size, void* d_ws, size_t ws_size,
                              hipStream_t stream) {
  const float* imgs  = (const float*)d_in[0];  // [64,3,384,384] f32
  const float* theta = (const float*)d_in[1];  // [64,2,3] f32
  float* out = (float*)d_out;                  // [64,3,384,384] f32

  // 384/16 = 24 tiles per dim; 4 waves (x-tiles) per 128-thread block.
  dim3 grid(24 / 4, 24, 64);
  dim3 block(128);
  hipLaunchKernelGGL(AffineTransforms_wmma_kernel, grid, block, 0, stream,
                     imgs, theta, out);
}